// NeuralNeighborhoodFlow_18287970746484
// MI455X (gfx1250) — compile-verified
//
#include <hip/hip_runtime.h>
#include <math.h>

// ---------------- problem constants (from reference setup_inputs) ------------
#define TT 5
#define BB 32
#define NN 128
#define DD 256
#define HH 512
#define MM (BB * NN)        // 4096 tangent rows
#define SUBSTEPS 2

// unified state: rows [0,MM) = Dy, rows [MM, MM+BB) = y, rest = zero padding
#define BM 128
#define MB 33               // ceil((MM+BB)/BM)
#define MP (MB * BM)        // 4224 padded rows

typedef __attribute__((ext_vector_type(2))) float v2f;
typedef __attribute__((ext_vector_type(8))) float v8f;

// ---------------- WMMA fp32 GEMM tiling --------------------------------------
#define BN 64
#define BK 32
#define LDT (BK + 2)        // padded LDS row stride -> conflict-free b64 reads

// ============================================================================
// GEMM1:  P[MP,H] = (S + c*dt*kS)[MP,D] @ W1[H,D]^T
//   rows < MM      : P = raw Jacobian pre-products (s-scaling deferred to GEMM2)
//   rows >= MM (y) : P = tanh(acc + b1) ; also emits sbuf = 1 - th^2
// ============================================================================
__global__ __launch_bounds__(256) void gemm1_kernel(
    const float* __restrict__ S, const float* __restrict__ kS,
    const float* __restrict__ W1, const float* __restrict__ b1,
    float* __restrict__ P, float* __restrict__ sbuf,
    const float* __restrict__ ts, int interval, float c)
{
    const int K = DD;          // 256
    const int N = HH;          // 512
    const float dt  = (ts[interval + 1] - ts[interval]) * (1.0f / SUBSTEPS);
    const float cdt = c * dt;

    const int bm  = blockIdx.x * BM;
    const int bn  = blockIdx.y * BN;
    const int tid = threadIdx.x;
    const int lane = tid & 31;
    const int wid  = tid >> 5;
    const int l15  = lane & 15;
    const int hi   = lane >> 4;
    const int wm   = wid >> 1;          // 0..3 (M)
    const int wn   = wid & 1;           // 0..1 (N)
    const int isY  = (bm >= MM);        // block-uniform region select

    __shared__ float As[BM * LDT];
    __shared__ float Bs[BN * LDT];

    v8f acc[2][2] = {};

    for (int k0 = 0; k0 < K; k0 += BK) {
        // ---- stage A tile (128 x 32) with fused RK4 AXPY (uniform over S) ----
        #pragma unroll
        for (int i = 0; i < 4; ++i) {
            int idx = tid + 256 * i;
            int row = idx >> 3, c4 = idx & 7;
            size_t g = (size_t)(bm + row) * K + k0 + c4 * 4;
            float4 v = *(const float4*)(S + g);
            if (c != 0.0f) {
                float4 kv = *(const float4*)(kS + g);
                v.x += cdt * kv.x; v.y += cdt * kv.y;
                v.z += cdt * kv.z; v.w += cdt * kv.w;
            }
            float* dst = &As[row * LDT + c4 * 4];
            ((float2*)dst)[0] = make_float2(v.x, v.y);
            ((float2*)dst)[1] = make_float2(v.z, v.w);
        }
        // ---- stage B tile (64 x 32) = rows of W1 ----
        #pragma unroll
        for (int i = 0; i < 2; ++i) {
            int idx = tid + 256 * i;
            int row = idx >> 3, c4 = idx & 7;
            float4 v = *(const float4*)(W1 + (size_t)(bn + row) * K + k0 + c4 * 4);
            float* dst = &Bs[row * LDT + c4 * 4];
            ((float2*)dst)[0] = make_float2(v.x, v.y);
            ((float2*)dst)[1] = make_float2(v.z, v.w);
        }
        __syncthreads();

        #pragma unroll
        for (int kk = 0; kk < BK; kk += 4) {
            v2f a[2], b[2];
            #pragma unroll
            for (int tm = 0; tm < 2; ++tm)
                a[tm] = *(const v2f*)&As[(wm * 32 + tm * 16 + l15) * LDT + kk + 2 * hi];
            #pragma unroll
            for (int tn = 0; tn < 2; ++tn)
                b[tn] = *(const v2f*)&Bs[(wn * 32 + tn * 16 + l15) * LDT + kk + 2 * hi];
            #pragma unroll
            for (int tm = 0; tm < 2; ++tm)
                #pragma unroll
                for (int tn = 0; tn < 2; ++tn)
                    acc[tm][tn] = __builtin_amdgcn_wmma_f32_16x16x4_f32(
                        false, a[tm], false, b[tn], (short)0, acc[tm][tn], false, false);
        }
        __syncthreads();
    }

    // ---- epilogue ----
    #pragma unroll
    for (int tm = 0; tm < 2; ++tm)
        #pragma unroll
        for (int tn = 0; tn < 2; ++tn) {
            int n = bn + wn * 32 + tn * 16 + l15;
            #pragma unroll
            for (int j = 0; j < 8; ++j) {
                int m = bm + wm * 32 + tm * 16 + j + 8 * hi;
                float val = acc[tm][tn][j];
                if (!isY) {
                    P[(size_t)m * N + n] = val;                     // raw Jv pre-product
                } else {
                    float t = tanhf(val + b1[n]);                   // y rows: hidden act
                    P[(size_t)m * N + n] = t;                       // th
                    if (m < MM + BB)
                        sbuf[(size_t)(m - MM) * HH + n] = 1.0f - t * t;  // sech^2
                }
            }
        }
}

// ============================================================================
// GEMM2:  out[MP,D] = A[MP,H] @ W2[D,H]^T
//   A rows < MM : P * s[b,h]  (deferred Jacobian scaling, fused in A-load)
//   A rows >= MM: P (= th)
//   epilogue    : RK4 k-accumulation; stage 3 fuses the state update + save
// ============================================================================
__global__ __launch_bounds__(256) void gemm2_kernel(
    const float* __restrict__ P, const float* __restrict__ W2,
    const float* __restrict__ b2, const float* __restrict__ sbuf,
    float* __restrict__ kS, float* __restrict__ ksum, float* __restrict__ S,
    const float* __restrict__ ts, int interval, int stage, float w,
    int save, int tindex, float* __restrict__ outYs, float* __restrict__ outDys)
{
    const int K = HH;          // 512
    const int N = DD;          // 256
    const float dt = (ts[interval + 1] - ts[interval]) * (1.0f / SUBSTEPS);
    const float s6 = dt * (1.0f / 6.0f);

    const int bm  = blockIdx.x * BM;
    const int bn  = blockIdx.y * BN;
    const int tid = threadIdx.x;
    const int lane = tid & 31;
    const int wid  = tid >> 5;
    const int l15  = lane & 15;
    const int hi   = lane >> 4;
    const int wm   = wid >> 1;
    const int wn   = wid & 1;
    const int isY  = (bm >= MM);

    __shared__ float As[BM * LDT];
    __shared__ float Bs[BN * LDT];

    v8f acc[2][2] = {};

    for (int k0 = 0; k0 < K; k0 += BK) {
        // ---- A tile: fused sech^2 scaling for tangent rows ----
        #pragma unroll
        for (int i = 0; i < 4; ++i) {
            int idx = tid + 256 * i;
            int row = idx >> 3, c4 = idx & 7;
            int r = bm + row;
            float4 v = *(const float4*)(P + (size_t)r * K + k0 + c4 * 4);
            if (!isY) {
                float4 sv = *(const float4*)(sbuf + (size_t)(r >> 7) * HH + k0 + c4 * 4);
                v.x *= sv.x; v.y *= sv.y; v.z *= sv.z; v.w *= sv.w;
            }
            float* dst = &As[row * LDT + c4 * 4];
            ((float2*)dst)[0] = make_float2(v.x, v.y);
            ((float2*)dst)[1] = make_float2(v.z, v.w);
        }
        // ---- B tile = rows of W2 ----
        #pragma unroll
        for (int i = 0; i < 2; ++i) {
            int idx = tid + 256 * i;
            int row = idx >> 3, c4 = idx & 7;
            float4 v = *(const float4*)(W2 + (size_t)(bn + row) * K + k0 + c4 * 4);
            float* dst = &Bs[row * LDT + c4 * 4];
            ((float2*)dst)[0] = make_float2(v.x, v.y);
            ((float2*)dst)[1] = make_float2(v.z, v.w);
        }
        __syncthreads();

        #pragma unroll
        for (int kk = 0; kk < BK; kk += 4) {
            v2f a[2], b[2];
            #pragma unroll
            for (int tm = 0; tm < 2; ++tm)
                a[tm] = *(const v2f*)&As[(wm * 32 + tm * 16 + l15) * LDT + kk + 2 * hi];
            #pragma unroll
            for (int tn = 0; tn < 2; ++tn)
                b[tn] = *(const v2f*)&Bs[(wn * 32 + tn * 16 + l15) * LDT + kk + 2 * hi];
            #pragma unroll
            for (int tm = 0; tm < 2; ++tm)
                #pragma unroll
                for (int tn = 0; tn < 2; ++tn)
                    acc[tm][tn] = __builtin_amdgcn_wmma_f32_16x16x4_f32(
                        false, a[tm], false, b[tn], (short)0, acc[tm][tn], false, false);
        }
        __syncthreads();
    }

    // ---- epilogue: RK4 accumulation / fused final update ----
    #pragma unroll
    for (int tm = 0; tm < 2; ++tm)
        #pragma unroll
        for (int tn = 0; tn < 2; ++tn) {
            int n = bn + wn * 32 + tn * 16 + l15;
            #pragma unroll
            for (int j = 0; j < 8; ++j) {
                int m = bm + wm * 32 + tm * 16 + j + 8 * hi;
                size_t o = (size_t)m * N + n;
                float val = acc[tm][tn][j];
                if (isY) val += b2[n];                 // dy gets its bias
                if (stage < 3) {
                    kS[o] = val;                        // next-stage input
                    if (stage == 0) ksum[o] = val;
                    else            ksum[o] += w * val;
                } else {
                    // S += dt/6 * (k1 + 2k2 + 2k3 + k4), optional save
                    float v = S[o] + s6 * (ksum[o] + val);
                    S[o] = v;
                    if (save) {
                        if (m < MM)
                            outDys[(size_t)tindex * MM * DD + o] = v;
                        else if (m < MM + BB)
                            outYs[(size_t)tindex * BB * DD + (size_t)(m - MM) * DD + n] = v;
                    }
                }
            }
        }
}

// ============================================================================
extern "C" void kernel_launch(void* const* d_in, const int* in_sizes, int n_in,
                              void* d_out, int out_size, void* d_ws, size_t ws_size,
                              hipStream_t stream)
{
    const float* ts  = (const float*)d_in[0];
    const float* y0  = (const float*)d_in[1];
    const float* Dy0 = (const float*)d_in[2];
    const float* W1  = (const float*)d_in[3];
    const float* b1  = (const float*)d_in[4];
    const float* W2  = (const float*)d_in[5];
    const float* b2  = (const float*)d_in[6];

    float* outYs  = (float*)d_out;                         // [T,B,D]
    float* outDys = (float*)d_out + (size_t)TT * BB * DD;  // [T,B,N,D]

    // workspace carve-out (all fp32)
    char* wsp = (char*)d_ws;
    auto alloc = [&](size_t nfloats) {
        float* p = (float*)wsp;
        wsp += nfloats * sizeof(float);
        return p;
    };
    float* S_cur = alloc((size_t)MP * DD);   // [Dy ; y ; 0-pad]
    float* kS    = alloc((size_t)MP * DD);   // current stage derivative
    float* ksum  = alloc((size_t)MP * DD);   // RK4 weighted sum
    float* P     = alloc((size_t)MP * HH);   // Jv pre-products / th rows
    float* sbuf  = alloc((size_t)BB * HH);   // sech^2

    // init state (capture-safe async ops); zero padding keeps replays finite
    hipMemsetAsync(S_cur, 0, (size_t)MP * DD * sizeof(float), stream);
    hipMemcpyAsync(S_cur, Dy0, (size_t)MM * DD * sizeof(float),
                   hipMemcpyDeviceToDevice, stream);
    hipMemcpyAsync(S_cur + (size_t)MM * DD, y0, (size_t)BB * DD * sizeof(float),
                   hipMemcpyDeviceToDevice, stream);
    hipMemcpyAsync(outYs,  y0,  (size_t)BB * DD * sizeof(float),
                   hipMemcpyDeviceToDevice, stream);
    hipMemcpyAsync(outDys, Dy0, (size_t)MM * DD * sizeof(float),
                   hipMemcpyDeviceToDevice, stream);

    const float cs[4] = {0.0f, 0.5f, 0.5f, 1.0f};  // RK4 stage offsets
    const float wg[4] = {1.0f, 2.0f, 2.0f, 1.0f};  // RK4 stage weights

    for (int interval = 0; interval < TT - 1; ++interval) {
        for (int sub = 0; sub < SUBSTEPS; ++sub) {
            for (int st = 0; st < 4; ++st) {
                gemm1_kernel<<<dim3(MB, HH / BN), 256, 0, stream>>>(
                    S_cur, kS, W1, b1, P, sbuf, ts, interval, cs[st]);
                int save = (st == 3 && sub == SUBSTEPS - 1) ? 1 : 0;
                gemm2_kernel<<<dim3(MB, DD / BN), 256, 0, stream>>>(
                    P, W2, b2, sbuf, kS, ksum, S_cur,
                    ts, interval, st, wg[st], save, interval + 1, outYs, outDys);
            }
        }
    }
}